// CapsuleLayer_39041252721228
// MI455X (gfx1250) — compile-verified
//
#include <hip/hip_runtime.h>
#include <hip/hip_bf16.h>

// CapsuleLayer on gfx1250:
//   B=16, I=1152, Din=16, N=32, D=64, ROUTINGS=3
// Kernel 1: x_hat[b,n,i,d] = sum_k W[n,i,d,k]*in[b,i,k] via V_WMMA_F32_16X16X4_F32
//           (per (n,i): [16b x 16k] @ [16k x 64d], one wave per (n,i))
// Kernel 2: dynamic routing per (b,n); logits are always x_hat*acc (acc = sum of
//           previous outputs), so only a 64-float acc is carried. x_hat slice
//           (294KB) stays L2-resident across the 6 passes.

typedef __attribute__((ext_vector_type(2))) float v2f;
typedef __attribute__((ext_vector_type(8))) float v8f;

#define BATCH 16
#define ICAPS 1152
#define DIN   16
#define NCAPS 32
#define DOUT  64
#define ROUTINGS 3

// ---------------------------------------------------------------------------
// Kernel 1: x_hat via f32 WMMA. One wave handles one (n,i) pair:
//   A = inputs[:, i, :]        (M=16 batch, K=16)
//   B = W[n,i,:,:]^T           (K=16, N=64 -> 4 tiles of 16)
//   D = x_hat[:, n, i, :]      (16 x 64)
// ISA layouts (7.12.2, 32-bit, wave32):
//   A 16x4 : lane&15 = M; VGPR0/1 = K {0,1} (lanes 0-15) or {2,3} (lanes 16-31)
//   B 4x16 : lane&15 = N; VGPR0/1 = K {0,1} (lanes 0-15) or {2,3} (lanes 16-31)
//   C/D    : VGPR r -> M = r + 8*(lane>=16), N = lane&15
// ---------------------------------------------------------------------------
__global__ __launch_bounds__(256) void capsule_xhat_wmma(
    const float* __restrict__ inp,   // [B, I, Din]
    const float* __restrict__ W,     // [N, I, D, Din]
    float* __restrict__ xhat)        // [B, N, I, D]
{
  const int wave = (blockIdx.x * 256 + threadIdx.x) >> 5;  // 0 .. N*I-1 exactly
  const int n    = wave / ICAPS;
  const int i    = wave - n * ICAPS;
  const int lane = threadIdx.x & 31;
  const int half = lane >> 4;    // 0: lanes 0-15, 1: lanes 16-31
  const int mrow = lane & 15;    // A: batch row / B: d column (within tile)

  const float* inp_base = inp + ((size_t)mrow * ICAPS + i) * DIN;       // b = mrow
  const float* w_base   = W   + ((size_t)n * ICAPS + i) * (DOUT * DIN); // [d][k]

  v8f acc0 = {}, acc1 = {}, acc2 = {}, acc3 = {};

#pragma unroll
  for (int kk = 0; kk < 4; ++kk) {
    const int ka = kk * 4 + half * 2;                 // K pair (ka, ka+1)
    v2f a  = *(const v2f*)(inp_base + ka);
    v2f b0 = *(const v2f*)(w_base + ( 0 + mrow) * DIN + ka);
    v2f b1 = *(const v2f*)(w_base + (16 + mrow) * DIN + ka);
    v2f b2 = *(const v2f*)(w_base + (32 + mrow) * DIN + ka);
    v2f b3 = *(const v2f*)(w_base + (48 + mrow) * DIN + ka);
    acc0 = __builtin_amdgcn_wmma_f32_16x16x4_f32(false, a, false, b0, (short)0, acc0, false, false);
    acc1 = __builtin_amdgcn_wmma_f32_16x16x4_f32(false, a, false, b1, (short)0, acc1, false, false);
    acc2 = __builtin_amdgcn_wmma_f32_16x16x4_f32(false, a, false, b2, (short)0, acc2, false, false);
    acc3 = __builtin_amdgcn_wmma_f32_16x16x4_f32(false, a, false, b3, (short)0, acc3, false, false);
  }

#pragma unroll
  for (int r = 0; r < 8; ++r) {
    const int b = r + half * 8;
    float* dst = xhat + (((size_t)b * NCAPS + n) * ICAPS + i) * DOUT + mrow;
    dst[ 0] = acc0[r];
    dst[16] = acc1[r];
    dst[32] = acc2[r];
    dst[48] = acc3[r];
  }
}

// ---------------------------------------------------------------------------
// Kernel 2: routing. Block = one (b,n). 256 threads = 64 d-lanes x 4 i-groups.
// Logits b[i,d] == x_hat[i,d] * acc[d]; softmax over i per d, weighted sum,
// elementwise squash (size-1 dim quirk of the reference).
// ---------------------------------------------------------------------------
__global__ __launch_bounds__(256) void capsule_routing(
    const float* __restrict__ xhat,  // [B, N, I, D]
    float* __restrict__ out)         // [B, N, D]
{
  const int blk = blockIdx.x;        // b*NCAPS + n
  const int t   = threadIdx.x;
  const int d   = t & 63;
  const int g   = t >> 6;            // 0..3
  const int CH  = ICAPS / 4;         // 288 i's per group

  const float* base = xhat + (size_t)blk * ICAPS * DOUT + d;
  const float* p    = base + (size_t)g * CH * DOUT;

  __shared__ float red[8 * 64];      // [0..3][d]: max/e-sum, [4..7][d]: e*x-sum

  float accd = 0.0f;                 // sum of previous outputs (this d)
  float outv = 0.0f;

  for (int it = 0; it < ROUTINGS; ++it) {
    // ---- pass 1: max over i of x_hat*acc (softmax stabilization) ----
    float m = -3.402823466e38f;
    for (int ii = 0; ii < CH; ++ii) {
      float v = p[ii * DOUT];
      m = fmaxf(m, v * accd);
    }
    red[g * 64 + d] = m;
    __syncthreads();
    m = fmaxf(fmaxf(red[d], red[64 + d]), fmaxf(red[128 + d], red[192 + d]));
    __syncthreads();

    // ---- pass 2: Z = sum exp, S = sum exp * x_hat ----
    float es = 0.0f, esx = 0.0f;
    for (int ii = 0; ii < CH; ++ii) {
      float v = p[ii * DOUT];
      float e = __expf(v * accd - m);
      es  += e;
      esx += e * v;
    }
    red[g * 64 + d]       = es;
    red[256 + g * 64 + d] = esx;
    __syncthreads();
    float Z = red[d] + red[64 + d] + red[128 + d] + red[192 + d];
    float S = red[256 + d] + red[320 + d] + red[384 + d] + red[448 + d];
    __syncthreads();

    float s  = S / Z;                       // sum_i softmax(b)_i * x_hat_i
    float sq = s * s;                       // squash on size-1 last dim
    outv = (sq / (1.0f + sq)) * s * rsqrtf(sq + 1e-9f);

    if (it < ROUTINGS - 1) accd += outv;    // b += x_hat * out  ==> acc += out
  }

  if (g == 0) out[(size_t)blk * DOUT + d] = outv;
}

// ---------------------------------------------------------------------------
extern "C" void kernel_launch(void* const* d_in, const int* in_sizes, int n_in,
                              void* d_out, int out_size, void* d_ws, size_t ws_size,
                              hipStream_t stream) {
  const float* inputs = (const float*)d_in[0];   // [16, 1152, 16] f32
  const float* W      = (const float*)d_in[1];   // [1, 32, 1152, 64, 16] f32
  float* out          = (float*)d_out;           // [16, 32, 64, 1] f32
  float* xhat         = (float*)d_ws;            // needs 16*32*1152*64*4 = 151 MB

  // Kernel 1: one wave per (n,i): 32*1152 = 36864 waves, 8 waves/block.
  const int waves  = NCAPS * ICAPS;
  const int blocks = waves / 8;                  // 4608
  hipLaunchKernelGGL(capsule_xhat_wmma, dim3(blocks), dim3(256), 0, stream,
                     inputs, W, xhat);

  // Kernel 2: one block per (b,n).
  hipLaunchKernelGGL(capsule_routing, dim3(BATCH * NCAPS), dim3(256), 0, stream,
                     xhat, out);
}